// ConnectedFilterLayerBySingleThreshold_25984552141121
// MI455X (gfx1250) — compile-verified
//
#include <hip/hip_runtime.h>

// ---------------------------------------------------------------------------
// ConnectedFilterLayerBySingleThreshold on MI455X (gfx1250)
//   phase 1: c[i] = sigmoid(attr[i]-thr)*delta[i], c[0]=0          (stream)
//   phase 2: 24x pointer jumping  v+=v[p]; p=p[p]   (L2-resident, packed v|p)
//   phase 3: out[j] = v[pix[j]]   (async-to-LDS staged index stream, gather v)
// Workspace layout: two ping-pong buffers of VP{float v; int p;}[N] (128 MB).
// ---------------------------------------------------------------------------

typedef int   v2i __attribute__((ext_vector_type(2)));
typedef int   v4i __attribute__((ext_vector_type(4)));
typedef float v4f __attribute__((ext_vector_type(4)));

struct VP { float v; int p; };   // 8 bytes, packed so one b64 gather fetches both

#define LDS_AS3(p) ((__attribute__((address_space(3))) void*)(p))

// ---------------- phase 1: init packed (c, parent) -------------------------
__global__ __launch_bounds__(256) void init_vp_kernel(
    const float* __restrict__ attr, const float* __restrict__ thrp,
    const float* __restrict__ delta, const int* __restrict__ parent,
    VP* __restrict__ vp, int n)
{
    const int t    = blockIdx.x * 256 + threadIdx.x;
    const int base = t * 4;
    const float thr = thrp[0];                 // uniform scalar load
    if (base + 3 < n) {
        // one-shot streams: non-temporal so they don't pollute L2
        v4f a = __builtin_nontemporal_load((const v4f*)attr   + t);
        v4f d = __builtin_nontemporal_load((const v4f*)delta  + t);
        v4i p = __builtin_nontemporal_load((const v4i*)parent + t);
        // sigmoid(a-thr)*d == d / (1 + exp(thr-a))
        float c0 = d.x / (1.0f + __expf(thr - a.x));
        float c1 = d.y / (1.0f + __expf(thr - a.y));
        float c2 = d.z / (1.0f + __expf(thr - a.z));
        float c3 = d.w / (1.0f + __expf(thr - a.w));
        if (base == 0) c0 = 0.0f;              // root contributes nothing
        v4i o0 = { __float_as_int(c0), p.x, __float_as_int(c1), p.y };
        v4i o1 = { __float_as_int(c2), p.z, __float_as_int(c3), p.w };
        v4i* dst = (v4i*)(vp + base);          // temporal: reused 24x from L2
        dst[0] = o0;
        dst[1] = o1;
    } else {
        for (int k = base; k < n; ++k) {
            float c = delta[k] / (1.0f + __expf(thr - attr[k]));
            if (k == 0) c = 0.0f;
            VP o; o.v = c; o.p = parent[k];
            vp[k] = o;
        }
    }
}

// ---------------- phase 2: one pointer-jumping round -----------------------
// v_new[i] = v[i] + v[p[i]];  p_new[i] = p[p[i]]   (reads only `in`)
__global__ __launch_bounds__(256) void jump_kernel(
    const VP* __restrict__ in, VP* __restrict__ out, int n)
{
    const int t = blockIdx.x * 256 + threadIdx.x;
    const int i = t * 2;
    if (i + 1 < n) {
        v4i me = *(const v4i*)(in + i);        // {v0,p0,v1,p1} sequential b128
        v2i g0 = *(const v2i*)(in + me.y);     // single b64 gather: (v[p0],p[p0])
        v2i g1 = *(const v2i*)(in + me.w);     // single b64 gather: (v[p1],p[p1])
        float nv0 = __int_as_float(me.x) + __int_as_float(g0.x);
        float nv1 = __int_as_float(me.z) + __int_as_float(g1.x);
        v4i res = { __float_as_int(nv0), g0.y, __float_as_int(nv1), g1.y };
        *(v4i*)(out + i) = res;                // temporal: stays hot in L2
    } else if (i < n) {
        v2i me = *(const v2i*)(in + i);
        v2i g  = *(const v2i*)(in + me.y);
        v2i res = { __float_as_int(__int_as_float(me.x) + __int_as_float(g.x)), g.y };
        *(v2i*)(out + i) = res;
    }
}

// ---------------- phase 3: per-pixel gather (async double-buffered) --------
// Each wave stages its next index tile into LDS with per-lane
// global_load_async_to_lds_b128 while gathering the current tile; ASYNCcnt is
// per-wave and async loads complete in order, and each lane reads back only
// the LDS slot it staged, so s_wait_asynccnt alone orders producer/consumer.
__global__ __launch_bounds__(256) void gather_async_kernel(
    const VP* __restrict__ vp, const int* __restrict__ idx,
    float* __restrict__ out, int nvec, int ntiles)
{
    __shared__ v4i stage[2][256];              // 8 KB double buffer
    const int tid    = threadIdx.x;
    const int stride = gridDim.x;
    const v4i* idx4  = (const v4i*)idx;

    const int t0 = blockIdx.x;
    if (t0 >= ntiles) return;

    {   // prologue: stage first tile
        int vi = t0 * 256 + tid; if (vi > nvec - 1) vi = nvec - 1;
        asm volatile("global_load_async_to_lds_b128 %0, %1, off"
                     :: "v"(LDS_AS3(&stage[0][tid])), "v"(idx4 + vi)
                     : "memory");
    }

    int bi = 0;
    for (int t = t0; t < ntiles; t += stride) {
        const int tn = t + stride;
        if (tn < ntiles) {
            // stage next tile into the other buffer
            int vi = tn * 256 + tid; if (vi > nvec - 1) vi = nvec - 1;
            asm volatile("global_load_async_to_lds_b128 %0, %1, off"
                         :: "v"(LDS_AS3(&stage[bi ^ 1][tid])), "v"(idx4 + vi)
                         : "memory");
            // warm L2 for the tile after next (global_prefetch_b8)
            const int t2 = tn + stride;
            if (t2 < ntiles) __builtin_prefetch(idx4 + (t2 * 256 + tid), 0, 0);
            asm volatile("s_wait_asynccnt 0x1" ::: "memory");  // current tile done
        } else {
            asm volatile("s_wait_asynccnt 0x0" ::: "memory");  // drain
        }

        const v4i id = stage[bi][tid];          // ds_load_b128, own slot
        const int vi = t * 256 + tid;
        if (vi < nvec) {
            v4f o;
            o.x = vp[id.x].v;                   // temporal b32 gathers: v table
            o.y = vp[id.y].v;                   // is L2-resident (32 MB of 192)
            o.z = vp[id.z].v;
            o.w = vp[id.w].v;
            __builtin_nontemporal_store(o, (v4f*)out + vi);     // NT stream out
        }
        bi ^= 1;
    }
}

__global__ __launch_bounds__(256) void gather_tail_kernel(
    const VP* __restrict__ vp, const int* __restrict__ idx,
    float* __restrict__ out, int start, int total)
{
    const int i = start + blockIdx.x * 256 + threadIdx.x;
    if (i < total) out[i] = vp[idx[i]].v;
}

// ---------------------------------------------------------------------------
extern "C" void kernel_launch(void* const* d_in, const int* in_sizes, int n_in,
                              void* d_out, int out_size, void* d_ws, size_t ws_size,
                              hipStream_t stream)
{
    const float* attr   = (const float*)d_in[0];   // attr_scaled_1d  (N)
    const float* thr    = (const float*)d_in[1];   // threshold_norm  (1)
    const float* delta  = (const float*)d_in[2];   // node_level_delta(N)
    const int*   parent = (const int*)  d_in[3];   // parent          (N)
    const int*   pix    = (const int*)  d_in[4];   // pixel_node_idx  (out_size)
    float*       out    = (float*)d_out;

    const int N = in_sizes[0];

    VP* buf0 = (VP*)d_ws;                 // ping
    VP* buf1 = buf0 + (size_t)N;          // pong   (2*N*8 bytes total)

    // phase 1: init buf0 with (c, parent), 4 nodes/thread
    {
        int threads = (N + 3) / 4;
        int blocks  = (threads + 255) / 256;
        init_vp_kernel<<<blocks, 256, 0, stream>>>(attr, thr, delta, parent, buf0, N);
    }

    // phase 2: 24 pointer-jumping rounds, ping-pong (final result lands in buf0)
    {
        int threads = (N + 1) / 2;
        int blocks  = (threads + 255) / 256;
        for (int t = 0; t < 24; ++t) {
            VP* src = (t & 1) ? buf1 : buf0;
            VP* dst = (t & 1) ? buf0 : buf1;
            jump_kernel<<<blocks, 256, 0, stream>>>(src, dst, N);
        }
    }

    // phase 3: gather to pixels, 4 pixels/thread, async-staged, + scalar tail
    {
        int nvec = out_size >> 2;
        int rem  = out_size & 3;
        if (nvec > 0) {
            int ntiles = (nvec + 255) / 256;
            int grid   = ntiles < 4096 ? ntiles : 4096;
            gather_async_kernel<<<grid, 256, 0, stream>>>(buf0, pix, out, nvec, ntiles);
        }
        if (rem > 0) {
            gather_tail_kernel<<<1, 256, 0, stream>>>(buf0, pix, out, nvec * 4, out_size);
        }
    }
}